// TrapezoidalDiscretization_14791867367989
// MI455X (gfx1250) — compile-verified
//
#include <hip/hip_runtime.h>
#include <hip/hip_bf16.h>
#include <math.h>

typedef __attribute__((ext_vector_type(2))) float v2f;
typedef __attribute__((ext_vector_type(8))) float v8f;

#define DDIM 32
#define RDIM 64
#define WAVES_PER_BLOCK 2

// One K=4 slice of V_WMMA_F32_16X16X4_F32 (D = A(16x4) * B(4x16) + C)
__device__ __forceinline__ v8f wmma4(v2f a, v2f b, v8f c) {
    return __builtin_amdgcn_wmma_f32_16x16x4_f32(
        /*neg_a=*/false, a, /*neg_b=*/false, b,
        /*c_mod=*/(short)0, c, /*reuse_a=*/false, /*reuse_b=*/false);
}

// 32x32x32 f32 GEMM on LDS operands via 2x2 tiles of v_wmma_f32_16x16x4_f32.
// AddIA / AddIB fuse "+Identity" into the operand loads (no materialized I+M^k).
template <bool AddIA, bool AddIB>
__device__ __forceinline__ void gemm32(const float* As, const float* Bs,
                                       v8f acc[2][2], int lane) {
    const int half = lane >> 4;   // which lane-half (K split 0,1 | 2,3)
    const int l15  = lane & 15;
#pragma unroll
    for (int mi = 0; mi < 2; ++mi)
#pragma unroll
        for (int ni = 0; ni < 2; ++ni)
            acc[mi][ni] = {};

#pragma unroll
    for (int kt = 0; kt < 8; ++kt) {          // K = 32 in slices of 4
        const int k0 = kt * 4 + half * 2;     // this lane-half's K pair
        v2f a[2], b[2];
#pragma unroll
        for (int mi = 0; mi < 2; ++mi) {      // A operand: 16x4, row per lane
            const int m = mi * 16 + l15;
            float x = As[m * DDIM + k0];
            float y = As[m * DDIM + k0 + 1];
            if (AddIA) {
                if (m == k0)     x += 1.0f;
                if (m == k0 + 1) y += 1.0f;
            }
            a[mi].x = x; a[mi].y = y;
        }
#pragma unroll
        for (int ni = 0; ni < 2; ++ni) {      // B operand: 4x16, row per VGPR
            const int n = ni * 16 + l15;
            float x = Bs[k0 * DDIM + n];
            float y = Bs[(k0 + 1) * DDIM + n];
            if (AddIB) {
                if (k0 == n)     x += 1.0f;
                if (k0 + 1 == n) y += 1.0f;
            }
            b[ni].x = x; b[ni].y = y;
        }
#pragma unroll
        for (int mi = 0; mi < 2; ++mi)
#pragma unroll
            for (int ni = 0; ni < 2; ++ni)
                acc[mi][ni] = wmma4(a[mi], b[ni], acc[mi][ni]);
    }
}

// Scatter D-matrix layout (ISA 7.12.2: VGPR r -> rows r / r+8 across lane halves)
// back to row-major 32x32.
__device__ __forceinline__ void store32(float* dst, const v8f acc[2][2], int lane) {
    const int half = lane >> 4;
    const int l15  = lane & 15;
#pragma unroll
    for (int mi = 0; mi < 2; ++mi)
#pragma unroll
        for (int ni = 0; ni < 2; ++ni)
#pragma unroll
            for (int r = 0; r < 8; ++r) {
                const int row = mi * 16 + r + half * 8;
                const int col = ni * 16 + l15;
                dst[row * DDIM + col] = acc[mi][ni][r];
            }
}

__global__ __launch_bounds__(32 * WAVES_PER_BLOCK)
void trapezoidal_cayley_kernel(const float* __restrict__ dt,
                               const float* __restrict__ A,
                               const float* __restrict__ W,
                               const float* __restrict__ bias,
                               float* __restrict__ out) {
    // Dispatch-constant operands, loaded once per block.
    __shared__ float A_s[DDIM * DDIM];        // 4 KB
    __shared__ float W_s[DDIM * RDIM];        // 8 KB
    __shared__ float b_s[DDIM];
    // Per-wave 32x32 scratch matrices.
    __shared__ float bufM [WAVES_PER_BLOCK][DDIM * DDIM]; // M
    __shared__ float bufP2[WAVES_PER_BLOCK][DDIM * DDIM]; // M^2, later M^8
    __shared__ float bufP4[WAVES_PER_BLOCK][DDIM * DDIM]; // M^4
    __shared__ float bufT [WAVES_PER_BLOCK][DDIM * DDIM]; // running product

    const int tid = threadIdx.x;
    for (int i = tid; i < DDIM * DDIM; i += 32 * WAVES_PER_BLOCK) A_s[i] = A[i];
    for (int i = tid; i < DDIM * RDIM; i += 32 * WAVES_PER_BLOCK) W_s[i] = W[i];
    if (tid < DDIM) b_s[tid] = bias[tid];
    __syncthreads();

    const int w    = tid >> 5;                // wave within block
    const int lane = tid & 31;
    const int site = blockIdx.x * WAVES_PER_BLOCK + w;   // grid sized exactly

    // --- dt_proj + softplus: lane d computes dtp[d] ---
    const float* dtp_in = dt + (size_t)site * RDIM;
    float z = b_s[lane];
#pragma unroll 4
    for (int r = 0; r < RDIM; ++r)
        z = fmaf(W_s[lane * RDIM + r], dtp_in[r], z);
    const float sp = (z > 20.0f) ? z : log1pf(expf(z));
    const float h  = 0.5f * sp;

    // --- M = (0.5*dtp) row-scaled A : lane writes its row ---
    float* M = bufM[w];
#pragma unroll 4
    for (int j = 0; j < DDIM; ++j)
        M[lane * DDIM + j] = h * A_s[lane * DDIM + j];
    __syncthreads();

    // X = (I-M)^{-1}(I+M) = (I+M) * (I+M)(I+M^2)(I+M^4)(I+M^8) + O(M^16)
    v8f acc[2][2];

    gemm32<false, false>(M, M, acc, lane);                 // P2 = M*M
    __syncthreads(); store32(bufP2[w], acc, lane); __syncthreads();

    gemm32<false, false>(bufP2[w], bufP2[w], acc, lane);   // P4 = P2*P2
    __syncthreads(); store32(bufP4[w], acc, lane); __syncthreads();

    gemm32<true, true>(M, bufP2[w], acc, lane);            // T1 = (I+M)(I+P2)
    __syncthreads(); store32(bufT[w], acc, lane); __syncthreads();

    gemm32<false, false>(bufP4[w], bufP4[w], acc, lane);   // P8 = P4*P4
    __syncthreads(); store32(bufP2[w], acc, lane); __syncthreads();

    gemm32<false, true>(bufT[w], bufP4[w], acc, lane);     // T2 = T1*(I+P4)
    __syncthreads(); store32(bufT[w], acc, lane); __syncthreads();

    gemm32<false, true>(bufT[w], bufP2[w], acc, lane);     // P  = T2*(I+P8)
    __syncthreads(); store32(bufT[w], acc, lane); __syncthreads();

    gemm32<true, false>(M, bufT[w], acc, lane);            // X  = (I+M)*P
    store32(out + (size_t)site * (DDIM * DDIM), acc, lane);
}

extern "C" void kernel_launch(void* const* d_in, const int* in_sizes, int n_in,
                              void* d_out, int out_size, void* d_ws, size_t ws_size,
                              hipStream_t stream) {
    const float* dt   = (const float*)d_in[0];  // [B,S,R]
    const float* A    = (const float*)d_in[1];  // [D,D]
    const float* W    = (const float*)d_in[2];  // [D,R]
    const float* bias = (const float*)d_in[3];  // [D]
    float* out = (float*)d_out;                 // [B,S,D,D]

    const int nsites = in_sizes[0] / RDIM;      // B*S = 16384
    const int blocks = nsites / WAVES_PER_BLOCK;
    trapezoidal_cayley_kernel<<<blocks, 32 * WAVES_PER_BLOCK, 0, stream>>>(
        dt, A, W, bias, out);
}